// Attention_65111704208056
// MI455X (gfx1250) — compile-verified
//
#include <hip/hip_runtime.h>
#include <hip/hip_bf16.h>

typedef __attribute__((ext_vector_type(2))) float v2f;
typedef __attribute__((ext_vector_type(8))) float v8f;

#define B_ 4
#define S_ 1024
#define H_ 16
#define HS_ 64
#define D_ 1024

// ---------------------------------------------------------------------------
// GEMM: Y = X[M,K] * W[K,N] + bias, fp32 WMMA 16x16x4.
// 64x64 tile per 256-thread block (8 waves, each wave 16 rows x 32 cols).
// mode 0: flat [M,N] output. mode 1: head-split [B,H,S,HS] output.
// blockIdx.z selects among up to 3 weight/bias/output sets (fused QKV).
// ---------------------------------------------------------------------------
__global__ __launch_bounds__(256) void gemm_wmma_kernel(
    const float* __restrict__ X,
    const float* W0, const float* W1, const float* W2,
    const float* Bi0, const float* Bi1, const float* Bi2,
    float* O0, float* O1, float* O2,
    int M, int N, int K, int mode)
{
    const float* W  = (blockIdx.z == 0) ? W0  : (blockIdx.z == 1) ? W1  : W2;
    const float* Bi = (blockIdx.z == 0) ? Bi0 : (blockIdx.z == 1) ? Bi1 : Bi2;
    float*       O  = (blockIdx.z == 0) ? O0  : (blockIdx.z == 1) ? O1  : O2;

    __shared__ float Xs[64][34];   // 64 rows x 32 k (pad 34: even, conflict-free)
    __shared__ float Ws[32][68];   // 32 k x 64 cols

    const int tid  = threadIdx.x;
    const int lane = tid & 31;
    const int wave = tid >> 5;
    const int m0 = blockIdx.y * 64;
    const int n0 = blockIdx.x * 64;
    const int wm = (wave & 3) * 16;   // wave's 16-row block
    const int wn = (wave >> 2) * 32;  // wave's 32-col block

    const int mi  = lane & 15;
    const int kb2 = (lane >> 4) * 2;  // lanes 0-15 -> K+0/K+1 ; lanes 16-31 -> K+2/K+3

    v8f acc0 = {}; v8f acc1 = {};

    for (int k0 = 0; k0 < K; k0 += 32) {
        for (int i = tid; i < 64 * 8; i += 256) {      // X tile 64x32 (float4 strips)
            int r = i >> 3, c = (i & 7) * 4;
            const float* src = X + (size_t)(m0 + r) * K + k0 + c;
            Xs[r][c + 0] = src[0]; Xs[r][c + 1] = src[1];
            Xs[r][c + 2] = src[2]; Xs[r][c + 3] = src[3];
        }
        for (int i = tid; i < 32 * 16; i += 256) {     // W tile 32x64
            int r = i >> 4, c = (i & 15) * 4;
            const float* src = W + (size_t)(k0 + r) * N + n0 + c;
            Ws[r][c + 0] = src[0]; Ws[r][c + 1] = src[1];
            Ws[r][c + 2] = src[2]; Ws[r][c + 3] = src[3];
        }
        // prefetch next k-tile while this one is consumed (global_prefetch_b8)
        if (k0 + 32 < K) {
            if (tid < 64)                                   // X: 64 rows x 128B
                __builtin_prefetch(X + (size_t)(m0 + tid) * K + k0 + 32, 0, 1);
            else if (tid < 128) {                           // W: 32 rows x 2 cachelines
                int r = tid - 64;
                __builtin_prefetch(W + (size_t)(k0 + 32 + (r >> 1)) * N + n0 + (r & 1) * 32, 0, 1);
            }
        }
        __syncthreads();

        for (int kk = 0; kk < 32; kk += 4) {
            int ka = kk + kb2;
            v2f a;  a.x  = Xs[wm + mi][ka];        a.y  = Xs[wm + mi][ka + 1];
            v2f b0; b0.x = Ws[ka][wn + mi];        b0.y = Ws[ka + 1][wn + mi];
            v2f b1; b1.x = Ws[ka][wn + 16 + mi];   b1.y = Ws[ka + 1][wn + 16 + mi];
            acc0 = __builtin_amdgcn_wmma_f32_16x16x4_f32(false, a, false, b0, (short)0, acc0, false, false);
            acc1 = __builtin_amdgcn_wmma_f32_16x16x4_f32(false, a, false, b1, (short)0, acc1, false, false);
        }
        __syncthreads();
    }

    // C/D layout: VGPR i -> (M = i, N = lane) lanes 0-15 ; (M = i+8, N = lane-16) lanes 16-31
    const int rbase = m0 + wm + (lane >> 4) * 8;
    for (int i = 0; i < 8; ++i) {
        int row = rbase + i;
        int c0 = n0 + wn + mi;
        int c1 = c0 + 16;
        float v0 = acc0[i] + Bi[c0];
        float v1 = acc1[i] + Bi[c1];
        if (mode == 0) {
            O[(size_t)row * N + c0] = v0;
            O[(size_t)row * N + c1] = v1;
        } else {
            int b = row >> 10, s = row & 1023;                 // row = b*S + s
            int h0 = c0 >> 6, hs0 = c0 & 63;
            int h1 = c1 >> 6, hs1 = c1 & 63;
            O[(((size_t)b * H_ + h0) * S_ + s) * HS_ + hs0] = v0;
            O[(((size_t)b * H_ + h1) * S_ + s) * HS_ + hs1] = v1;
        }
    }
}

// ---------------------------------------------------------------------------
// Attention core: one block = 4 waves, one (b,h) and a 16-row q tile.
// Phase 1: scores = Q*K^T/8 + mask*(-1e4)   (WMMA, per-wave 256-col strip)
// Phase 2: row softmax (shfl_xor reductions), NT-store attn to d_out
// Phase 3: ctx = attn*V (WMMA, double-buffered V tiles), write [B,S,D] ctx
// ---------------------------------------------------------------------------
__global__ __launch_bounds__(128) void attn_kernel(
    const float* __restrict__ Qg, const float* __restrict__ Kg,
    const float* __restrict__ Vg, const int* __restrict__ mask,
    float* __restrict__ attn_out, float* __restrict__ ctx)
{
    extern __shared__ float smem[];
    const int SST = 1028;                       // padded score stride
    float* Qs  = smem;                          // 16*66
    float* Ks  = Qs + 16 * 66;                  // 4 * 16*66 (per-wave staging)
    float* Vs0 = Ks + 4 * 16 * 66;              // 16*66 (double buffer A)
    float* Vs1 = Vs0 + 16 * 66;                 // 16*66 (double buffer B)
    float* Ss  = Vs1 + 16 * 66;                 // 16*1028

    const int tid  = threadIdx.x;
    const int lane = tid & 31;
    const int wave = tid >> 5;
    const int q0 = blockIdx.x * 16;
    const int bh = blockIdx.y;                  // b*H + h
    const int b  = bh >> 4;
    const int h  = bh & 15;

    const int mi  = lane & 15;
    const int kb2 = (lane >> 4) * 2;

    // Q tile 16x64 cooperative load
    for (int i = tid; i < 16 * 64; i += 128) {
        int r = i >> 6, c = i & 63;
        Qs[r * 66 + c] = Qg[((size_t)bh * S_ + q0 + r) * HS_ + c];
    }
    __syncthreads();

    // ---- Phase 1: scores ----
    float* Ksw = Ks + wave * 16 * 66;
    for (int nt = 0; nt < 16; ++nt) {
        int n0c = wave * 256 + nt * 16;
        for (int i = lane; i < 16 * 64; i += 32) {       // stage 16 key rows
            int r = i >> 6, c = i & 63;
            Ksw[r * 66 + c] = Kg[((size_t)bh * S_ + n0c + r) * HS_ + c];
        }
        if (nt < 15)                                      // prefetch next K tile (4KB)
            __builtin_prefetch(Kg + ((size_t)bh * S_ + n0c + 16) * HS_ + lane * 32, 0, 1);
        v8f acc = {};
        for (int kk = 0; kk < HS_; kk += 4) {
            int ka = kk + kb2;
            v2f a; a.x = Qs[mi * 66 + ka];  a.y = Qs[mi * 66 + ka + 1];
            // B[ka][n] = K[n][ka] : transposed read == same pattern as A
            v2f bb; bb.x = Ksw[mi * 66 + ka]; bb.y = Ksw[mi * 66 + ka + 1];
            acc = __builtin_amdgcn_wmma_f32_16x16x4_f32(false, a, false, bb, (short)0, acc, false, false);
        }
        int n = n0c + mi;
        float mv = (float)mask[b * S_ + n] * -10000.0f;
        int rb = (lane >> 4) * 8;
        for (int i = 0; i < 8; ++i)
            Ss[(rb + i) * SST + n] = acc[i] * 0.125f + mv;   // 1/sqrt(64)
    }
    __syncthreads();

    // ---- Phase 2: softmax over each of the 16 rows ----
    for (int r = wave; r < 16; r += 4) {
        float mx = -3.4e38f;
        for (int j = lane; j < S_; j += 32) mx = fmaxf(mx, Ss[r * SST + j]);
        for (int off = 16; off > 0; off >>= 1) mx = fmaxf(mx, __shfl_xor(mx, off, 32));
        float sum = 0.0f;
        for (int j = lane; j < S_; j += 32) {
            float e = __expf(Ss[r * SST + j] - mx);
            Ss[r * SST + j] = e;
            sum += e;
        }
        for (int off = 16; off > 0; off >>= 1) sum += __shfl_xor(sum, off, 32);
        float inv = 1.0f / sum;
        size_t obase = ((size_t)bh * S_ + q0 + r) * S_;
        for (int j = lane; j < S_; j += 32) {
            float a = Ss[r * SST + j] * inv;
            Ss[r * SST + j] = a;
            __builtin_nontemporal_store(a, attn_out + obase + j);  // write-once: NT
        }
    }
    __syncthreads();

    // ---- Phase 3: ctx = attn * V, double-buffered V tiles ----
    v8f acc = {};
    const int c0 = wave * 16;
    for (int i = tid; i < 16 * 64; i += 128) {           // preload tile 0
        int r = i >> 6, c = i & 63;
        Vs0[r * 66 + c] = Vg[((size_t)bh * S_ + r) * HS_ + c];
    }
    for (int kt = 0; kt < 64; ++kt) {
        __syncthreads();                                  // cur tile visible to all
        float* cur = (kt & 1) ? Vs1 : Vs0;
        float* nxt = (kt & 1) ? Vs0 : Vs1;
        if (kt < 63) {                                    // overlap next tile load
            for (int i = tid; i < 16 * 64; i += 128) {
                int r = i >> 6, c = i & 63;
                nxt[r * 66 + c] = Vg[((size_t)bh * S_ + (kt + 1) * 16 + r) * HS_ + c];
            }
            if (kt < 62)                                  // prefetch tile kt+2 (4KB)
                __builtin_prefetch(Vg + ((size_t)bh * S_ + (kt + 2) * 16) * HS_ + lane * 32, 0, 1);
        }
        for (int kk = 0; kk < 16; kk += 4) {
            int ka = kk + kb2;
            v2f a;  a.x  = Ss[mi * SST + kt * 16 + ka]; a.y = Ss[mi * SST + kt * 16 + ka + 1];
            v2f bb; bb.x = cur[ka * 66 + c0 + mi];      bb.y = cur[(ka + 1) * 66 + c0 + mi];
            acc = __builtin_amdgcn_wmma_f32_16x16x4_f32(false, a, false, bb, (short)0, acc, false, false);
        }
    }
    int rb = (lane >> 4) * 8;
    for (int i = 0; i < 8; ++i) {
        int s = q0 + rb + i;
        ctx[((size_t)b * S_ + s) * D_ + h * HS_ + c0 + mi] = acc[i];
    }
}

// ---------------------------------------------------------------------------
// Residual add + LayerNorm: one 256-thread block per row of 1024.
// ---------------------------------------------------------------------------
__global__ __launch_bounds__(256) void ln_kernel(
    const float* __restrict__ x, const float* __restrict__ proj,
    const float* __restrict__ gamma, const float* __restrict__ beta,
    float* __restrict__ out)
{
    __shared__ float w1[8], w2[8];
    const int row = blockIdx.x;
    const int tid = threadIdx.x;
    const int lane = tid & 31, wave = tid >> 5;

    float v[4];
    float s1 = 0.0f, s2 = 0.0f;
    for (int j = 0; j < 4; ++j) {
        int c = tid + j * 256;
        float t = x[(size_t)row * D_ + c] + proj[(size_t)row * D_ + c];
        v[j] = t; s1 += t; s2 += t * t;
    }
    for (int off = 16; off > 0; off >>= 1) {
        s1 += __shfl_xor(s1, off, 32);
        s2 += __shfl_xor(s2, off, 32);
    }
    if (lane == 0) { w1[wave] = s1; w2[wave] = s2; }
    __syncthreads();
    float t1 = 0.0f, t2 = 0.0f;
    for (int i = 0; i < 8; ++i) { t1 += w1[i]; t2 += w2[i]; }
    float mu   = t1 * (1.0f / D_);
    float var  = t2 * (1.0f / D_) - mu * mu;
    float rstd = rsqrtf(var + 1e-6f);
    for (int j = 0; j < 4; ++j) {
        int c = tid + j * 256;
        float o = (v[j] - mu) * rstd * gamma[c] + beta[c];
        __builtin_nontemporal_store(o, out + (size_t)row * D_ + c);  // write-once: NT
    }
}

// ---------------------------------------------------------------------------
extern "C" void kernel_launch(void* const* d_in, const int* in_sizes, int n_in,
                              void* d_out, int out_size, void* d_ws, size_t ws_size,
                              hipStream_t stream)
{
    const float* x    = (const float*)d_in[0];
    const int*   mask = (const int*)  d_in[1];
    const float* wq   = (const float*)d_in[2];
    const float* bq   = (const float*)d_in[3];
    const float* wk   = (const float*)d_in[4];
    const float* bk   = (const float*)d_in[5];
    const float* wv   = (const float*)d_in[6];
    const float* bv   = (const float*)d_in[7];
    const float* wo   = (const float*)d_in[8];
    const float* bo   = (const float*)d_in[9];
    const float* gamma= (const float*)d_in[10];
    const float* beta = (const float*)d_in[11];

    float* out  = (float*)d_out;
    float* attn = out + (size_t)B_ * S_ * D_;      // tuple: (out, attn) concatenated

    const size_t TENS = (size_t)B_ * S_ * D_;      // 4.19M floats
    float* ws   = (float*)d_ws;
    float* Q    = ws;
    float* K    = ws + TENS;
    float* V    = ws + 2 * TENS;
    float* ctx  = ws + 3 * TENS;
    float* proj = Q;                               // Q dead after attn; reuse

    const int M = B_ * S_;                         // 4096

    // 1) fused QKV projections, head-split outputs
    gemm_wmma_kernel<<<dim3(D_ / 64, M / 64, 3), 256, 0, stream>>>(
        x, wq, wk, wv, bq, bk, bv, Q, K, V, M, D_, D_, /*mode=*/1);

    // 2) attention core
    size_t smem = (size_t)(16 * 66 + 4 * 16 * 66 + 2 * 16 * 66 + 16 * 1028) * sizeof(float);
    attn_kernel<<<dim3(S_ / 16, B_ * H_), 128, smem, stream>>>(
        Q, K, V, mask, attn, ctx);

    // 3) output projection (flat layout)
    gemm_wmma_kernel<<<dim3(D_ / 64, M / 64, 1), 256, 0, stream>>>(
        ctx, wo, wo, wo, bo, bo, bo, proj, proj, proj, M, D_, D_, /*mode=*/0);

    // 4) residual + layernorm
    ln_kernel<<<M, 256, 0, stream>>>(x, proj, gamma, beta, out);
}